// GATModule_52123723105127
// MI455X (gfx1250) — compile-verified
//
#include <hip/hip_runtime.h>
#include <math.h>

#define N_NODES 50000
#define N_EDGES 800000
#define DIM     128
#define HEADS   4
#define CHEAD   32
#define E2      (N_EDGES + N_NODES)     // 850000 (self-loops appended)
#define NEG_SLOPE 0.2f
#define LN_EPS    1e-5f
#define NORM_EPS  1e-12f

typedef __attribute__((ext_vector_type(2))) float v2f;
typedef __attribute__((ext_vector_type(8))) float v8f;

__device__ __forceinline__ float waveRedSum(float v) {
#pragma unroll
  for (int o = 16; o > 0; o >>= 1) v += __shfl_xor(v, o, 32);
  return v;
}

// sign-split float atomic max (init must be -inf)
__device__ __forceinline__ void atomicMaxFloat(float* addr, float val) {
  if (val >= 0.0f) atomicMax((int*)addr, __float_as_int(val));
  else             atomicMin((unsigned int*)addr, __float_as_uint(val));
}

// ---------------- Kernel 1: LayerNorm -> ReLU, plus ||h||2 ----------------
__global__ __launch_bounds__(256)
void ln_relu_kernel(const float* __restrict__ x, const float* __restrict__ gamma,
                    const float* __restrict__ beta, float* __restrict__ h,
                    float* __restrict__ hnorm) {
  int node = blockIdx.x * 8 + (threadIdx.x >> 5);
  int lane = threadIdx.x & 31;
  if (node >= N_NODES) return;
  const float4 xv = ((const float4*)x)[node * 32 + lane];
  float mean = waveRedSum(xv.x + xv.y + xv.z + xv.w) * (1.0f / 128.0f);
  float dx = xv.x - mean, dy = xv.y - mean, dz = xv.z - mean, dw = xv.w - mean;
  float var = waveRedSum(dx * dx + dy * dy + dz * dz + dw * dw) * (1.0f / 128.0f);
  float rstd = rsqrtf(var + LN_EPS);
  const float4 g = ((const float4*)gamma)[lane];
  const float4 b = ((const float4*)beta)[lane];
  float4 hv;
  hv.x = fmaxf(dx * rstd * g.x + b.x, 0.0f);
  hv.y = fmaxf(dy * rstd * g.y + b.y, 0.0f);
  hv.z = fmaxf(dz * rstd * g.z + b.z, 0.0f);
  hv.w = fmaxf(dw * rstd * g.w + b.w, 0.0f);
  float ss = waveRedSum(hv.x * hv.x + hv.y * hv.y + hv.z * hv.z + hv.w * hv.w);
  if (lane == 0) hnorm[node] = sqrtf(ss);
  ((float4*)h)[node * 32 + lane] = hv;
}

// ---- Kernel 2a: pre-pack Wl/Wr into per-lane WMMA B-fragment order ----
// packed[((ct*32 + kk)*32 + lane)] = {Wl[k0,c], Wl[k0+1,c], Wr[k0,c], Wr[k0+1,c]}
// with k0 = kk*4 + 2*(lane>>4), c = ct*16 + (lane&15)  -> one b128 per lane per K-step
__global__ __launch_bounds__(256)
void pack_w_kernel(const float* __restrict__ Wl, const float* __restrict__ Wr,
                   float4* __restrict__ packed) {
  int idx = blockIdx.x * blockDim.x + threadIdx.x;   // 8 * 32 * 32 = 8192
  if (idx >= 8 * 32 * 32) return;
  int lane = idx & 31;
  int kk   = (idx >> 5) & 31;
  int ct   = idx >> 10;
  int k0   = kk * 4 + 2 * (lane >> 4);
  int c    = ct * 16 + (lane & 15);
  float4 v;
  v.x = Wl[k0 * DIM + c];
  v.y = Wl[(k0 + 1) * DIM + c];
  v.z = Wr[k0 * DIM + c];
  v.w = Wr[(k0 + 1) * DIM + c];
  packed[idx] = v;
}

// ---- Kernel 2b: dual GEMM xl = h@Wl + bl, xr = h@Wr + br via V_WMMA_F32_16X16X4_F32 ----
__global__ __launch_bounds__(256)
void gemm_dual_kernel(const float* __restrict__ h, const float4* __restrict__ packedW,
                      const float* __restrict__ bl, const float* __restrict__ br,
                      float* __restrict__ xl, float* __restrict__ xr) {
  __shared__ float As[16 * 132];                 // 16x128 strip, stride 132 (bank-conflict pad)
  const int row0 = blockIdx.x * 16;
  {
    int row = threadIdx.x >> 4;                  // 256 threads x 8 floats = 2048
    int col = (threadIdx.x & 15) * 8;
    const float4* s = (const float4*)(h + (row0 + row) * DIM + col);
    float4 v0 = s[0], v1 = s[1];
    float4* d = (float4*)(As + row * 132 + col);
    d[0] = v0; d[1] = v1;
  }
  __syncthreads();

  const int lane  = threadIdx.x & 31;
  const int ct    = threadIdx.x >> 5;            // wave -> 16-col tile
  const int col0  = ct * 16;
  const int m     = lane & 15;
  const int khalf = lane >> 4;                   // A/B: lanes 0-15 hold K 0,1; 16-31 hold K 2,3
  const float4* pw = packedW + (size_t)ct * 32 * 32 + lane;
  v8f accL = {}; v8f accR = {};
#pragma unroll
  for (int kk = 0; kk < 32; ++kk) {
    int k0 = kk * 4 + 2 * khalf;
    v2f a;  a.x = As[m * 132 + k0]; a.y = As[m * 132 + k0 + 1];
    float4 bv = pw[kk * 32];                     // single coalesced global_load_b128
    v2f b0; b0.x = bv.x; b0.y = bv.y;
    v2f b1; b1.x = bv.z; b1.y = bv.w;
    accL = __builtin_amdgcn_wmma_f32_16x16x4_f32(false, a, false, b0, (short)0, accL, false, false);
    accR = __builtin_amdgcn_wmma_f32_16x16x4_f32(false, a, false, b1, (short)0, accR, false, false);
  }
  // C/D layout: vgpr i -> row i (lanes 0-15) / row i+8 (lanes 16-31), N = lane&15
  const int nc = col0 + (lane & 15);
  const float bL = bl[nc], bR = br[nc];
#pragma unroll
  for (int i = 0; i < 8; ++i) {
    int row = row0 + i + khalf * 8;
    xl[row * DIM + nc] = accL[i] + bL;
    xr[row * DIM + nc] = accR[i] + bR;
  }
}

// ---------------- Kernel 3: init accumulators ----------------
__global__ __launch_bounds__(256)
void init_kernel(float* __restrict__ acc, float* __restrict__ mmax, float* __restrict__ denom) {
  int i = blockIdx.x * blockDim.x + threadIdx.x;
  if (i < N_NODES * DIM) acc[i] = 0.0f;
  if (i < N_NODES * HEADS) { mmax[i] = -INFINITY; denom[i] = 0.0f; }
}

// ---------------- Kernel 4: per-edge GATv2 scores + segment max ----------------
__global__ __launch_bounds__(256)
void score_kernel(const int* __restrict__ ei, const float* __restrict__ xl,
                  const float* __restrict__ xr, const float* __restrict__ att,
                  float* __restrict__ ebuf, float* __restrict__ mmax) {
  int e = blockIdx.x * 8 + (threadIdx.x >> 5);
  if (e >= E2) return;
  int lane = threadIdx.x & 31;
  int src, dst;
  if (e < N_EDGES) { src = ei[e]; dst = ei[N_EDGES + e]; }
  else             { src = dst = e - N_EDGES; }               // self loops
  const float4 av = ((const float4*)att)[lane];
  const float4 lv = ((const float4*)xl)[src * 32 + lane];
  const float4 rv = ((const float4*)xr)[dst * 32 + lane];
  float t, p = 0.0f;
  t = lv.x + rv.x; t = t > 0.0f ? t : NEG_SLOPE * t; p += t * av.x;
  t = lv.y + rv.y; t = t > 0.0f ? t : NEG_SLOPE * t; p += t * av.y;
  t = lv.z + rv.z; t = t > 0.0f ? t : NEG_SLOPE * t; p += t * av.z;
  t = lv.w + rv.w; t = t > 0.0f ? t : NEG_SLOPE * t; p += t * av.w;
  p += __shfl_xor(p, 1, 32);                                  // reduce 8-lane head group
  p += __shfl_xor(p, 2, 32);
  p += __shfl_xor(p, 4, 32);
  if ((lane & 7) == 0) {
    int head = lane >> 3;
    ebuf[e * HEADS + head] = p;
    atomicMaxFloat(&mmax[dst * HEADS + head], p);
  }
}

// ---------------- Kernel 5: p = exp(e - m[dst]); denom[dst] += p ----------------
__global__ __launch_bounds__(256)
void exp_denom_kernel(const int* __restrict__ ei, float* __restrict__ ebuf,
                      const float* __restrict__ mmax, float* __restrict__ denom) {
  int idx = blockIdx.x * blockDim.x + threadIdx.x;
  if (idx >= E2 * HEADS) return;
  int e = idx >> 2, hh = idx & 3;
  int dst = (e < N_EDGES) ? ei[N_EDGES + e] : e - N_EDGES;
  float p = __expf(ebuf[idx] - mmax[dst * HEADS + hh]);
  ebuf[idx] = p;
  unsafeAtomicAdd(&denom[dst * HEADS + hh], p);
}

// ---------------- Kernel 6: out[dst] += alpha * xl[src] ----------------
__global__ __launch_bounds__(256)
void aggregate_kernel(const int* __restrict__ ei, const float* __restrict__ ebuf,
                      const float* __restrict__ denom, const float* __restrict__ xl,
                      float* __restrict__ acc) {
  int e = blockIdx.x * 8 + (threadIdx.x >> 5);
  if (e >= E2) return;
  int lane = threadIdx.x & 31;
  int src, dst;
  if (e < N_EDGES) { src = ei[e]; dst = ei[N_EDGES + e]; }
  else             { src = dst = e - N_EDGES; }
  int head = lane >> 3;
  float alpha = ebuf[e * HEADS + head] / denom[dst * HEADS + head];
  const float4 lv = ((const float4*)xl)[src * 32 + lane];
  float* base = acc + dst * DIM + lane * 4;
  unsafeAtomicAdd(base + 0, alpha * lv.x);
  unsafeAtomicAdd(base + 1, alpha * lv.y);
  unsafeAtomicAdd(base + 2, alpha * lv.z);
  unsafeAtomicAdd(base + 3, alpha * lv.w);
}

// ---------------- Kernel 7: +bias, MsgNorm, residual (in place on d_out) ----------------
__global__ __launch_bounds__(256)
void finalize_kernel(const float* __restrict__ x, const float* __restrict__ bias,
                     const float* __restrict__ hnorm, const float* __restrict__ scale,
                     float* __restrict__ out) {
  int node = blockIdx.x * 8 + (threadIdx.x >> 5);
  int lane = threadIdx.x & 31;
  if (node >= N_NODES) return;
  float4 a = ((float4*)out)[node * 32 + lane];
  const float4 b = ((const float4*)bias)[lane];
  a.x += b.x; a.y += b.y; a.z += b.z; a.w += b.w;
  float ss = waveRedSum(a.x * a.x + a.y * a.y + a.z * a.z + a.w * a.w);
  float f = hnorm[node] * scale[0] / fmaxf(sqrtf(ss), NORM_EPS);
  const float4 xv = ((const float4*)x)[node * 32 + lane];
  float4 r;
  r.x = xv.x + a.x * f; r.y = xv.y + a.y * f;
  r.z = xv.z + a.z * f; r.w = xv.w + a.w * f;
  ((float4*)out)[node * 32 + lane] = r;
}

extern "C" void kernel_launch(void* const* d_in, const int* in_sizes, int n_in,
                              void* d_out, int out_size, void* d_ws, size_t ws_size,
                              hipStream_t stream) {
  const float* x     = (const float*)d_in[0];
  const int*   ei    = (const int*)d_in[1];
  const float* gamma = (const float*)d_in[2];
  const float* beta  = (const float*)d_in[3];
  const float* Wl    = (const float*)d_in[4];
  const float* bl    = (const float*)d_in[5];
  const float* Wr    = (const float*)d_in[6];
  const float* br    = (const float*)d_in[7];
  const float* att   = (const float*)d_in[8];
  const float* bias  = (const float*)d_in[9];
  const float* scale = (const float*)d_in[10];
  float* out = (float*)d_out;

  float* ws = (float*)d_ws;
  float* h     = ws;                               // N*128
  float* hnorm = h + (size_t)N_NODES * DIM;        // N
  float* xl    = hnorm + N_NODES;                  // N*128
  float* xr    = xl + (size_t)N_NODES * DIM;       // N*128
  float* ebuf  = xr + (size_t)N_NODES * DIM;       // E2*H
  float* mmax  = ebuf + (size_t)E2 * HEADS;        // N*H
  float* denom = mmax + (size_t)N_NODES * HEADS;   // N*H
  float4* packedW = (float4*)(denom + (size_t)N_NODES * HEADS);  // 8192 float4 (128KB)

  pack_w_kernel<<<(8 * 32 * 32 + 255) / 256, 256, 0, stream>>>(Wl, Wr, packedW);
  ln_relu_kernel<<<(N_NODES + 7) / 8, 256, 0, stream>>>(x, gamma, beta, h, hnorm);
  gemm_dual_kernel<<<N_NODES / 16, 256, 0, stream>>>(h, packedW, bl, br, xl, xr);
  init_kernel<<<(N_NODES * DIM + 255) / 256, 256, 0, stream>>>(out, mmax, denom);
  score_kernel<<<(E2 + 7) / 8, 256, 0, stream>>>(ei, xl, xr, att, ebuf, mmax);
  exp_denom_kernel<<<(E2 * HEADS + 255) / 256, 256, 0, stream>>>(ei, ebuf, mmax, denom);
  aggregate_kernel<<<(E2 + 7) / 8, 256, 0, stream>>>(ei, ebuf, denom, xl, out);
  finalize_kernel<<<(N_NODES + 7) / 8, 256, 0, stream>>>(x, bias, hnorm, scale, out);
}